// Net_18734647345152
// MI455X (gfx1250) — compile-verified
//
#include <hip/hip_runtime.h>

// Scatter-add: out = A; for i in [0, N_UPD): out[index[i], :] += B[i, :]
// D = 128 fp32 per row. Bandwidth-bound (~0.62 GB total HBM traffic).
// Strategy: out (51.2 MB) lives in the 192 MB L2; all fp32 atomics resolve
// in L2. B (512 MB) is streamed once with nontemporal loads so it does not
// evict the accumulator working set.

typedef float v4f __attribute__((ext_vector_type(4)));

#define FEAT 128
#define FEAT4 (FEAT / 4)

// One wave32 per update row; lane L covers columns [4L, 4L+4).
// Indices are fetched 32-at-a-time with one coalesced load per wave, then
// broadcast per-row via v_readlane_b32 (uniform lane select -> SGPR base for
// the atomics: saddr + lane*16 voffset + imm 0/4/8/12).
__global__ void scatter_add(const int* __restrict__ index,
                            const v4f* __restrict__ B,
                            float* __restrict__ out,
                            int n_upd, int rows_per_wave) {
    const int lane = threadIdx.x & 31;
    const int wave = (blockIdx.x * blockDim.x + threadIdx.x) >> 5;

    int r0 = wave * rows_per_wave;
    int r1 = r0 + rows_per_wave;
    if (r1 > n_upd) r1 = n_upd;

    for (int base = r0; base < r1; base += 32) {
        int cnt = r1 - base;
        if (cnt > 32) cnt = 32;

        // One coalesced load of 32 consecutive indices for this wave.
        int my_idx = 0;
        if (lane < cnt) my_idx = index[base + lane];

        for (int j = 0; j < cnt; ++j) {
            const int row = base + j;
            // Uniform lane select -> v_readlane_b32 -> SGPR row index.
            const int idx = __builtin_amdgcn_readlane(my_idx, j);

            // Stream one float4 of B per lane: coalesced 512 B per row,
            // NT hint (read-once data must not displace out's L2 set).
            v4f b = __builtin_nontemporal_load(B + (size_t)row * FEAT4 + lane);

            float* dst = out + (size_t)idx * FEAT + lane * 4;
            // Native fp32 atomic add (global_atomic_add_f32, no CAS loop).
            // Result discarded -> non-returning atomic (STOREcnt).
            unsafeAtomicAdd(dst + 0, b.x);
            unsafeAtomicAdd(dst + 1, b.y);
            unsafeAtomicAdd(dst + 2, b.z);
            unsafeAtomicAdd(dst + 3, b.w);
        }
    }
}

// out[0..n4) = A[0..n4), float4-vectorized. NT load (A read exactly once),
// regular store (leave out dirty in L2 for the atomic phase).
__global__ void copy_rows(const v4f* __restrict__ A, v4f* __restrict__ out, int n4) {
    int i = blockIdx.x * blockDim.x + threadIdx.x;
    if (i < n4) {
        v4f v = __builtin_nontemporal_load(A + i);
        out[i] = v;
    }
}

extern "C" void kernel_launch(void* const* d_in, const int* in_sizes, int n_in,
                              void* d_out, int out_size, void* d_ws, size_t ws_size,
                              hipStream_t stream) {
    const int*   index = (const int*)d_in[0];       // (N_UPD,) int32
    const float* A     = (const float*)d_in[1];     // (N_ROWS, 128) fp32
    const v4f*   B     = (const v4f*)d_in[2];       // (N_UPD, 128) fp32 as float4
    float*       out   = (float*)d_out;             // (N_ROWS, 128) fp32

    const int n_upd   = in_sizes[0];
    const int n_elems = in_sizes[1];                // N_ROWS * 128
    const int n4      = n_elems / 4;

    // Phase 1: out = A
    {
        const int tpb = 256;
        const int blocks = (n4 + tpb - 1) / tpb;
        copy_rows<<<blocks, tpb, 0, stream>>>((const v4f*)A, (v4f*)out, n4);
    }

    // Phase 2: atomic scatter of B into out
    {
        const int tpb = 256;                        // 8 wave32 per block
        const int blocks = 4096;
        const int n_waves = blocks * (tpb / 32);    // 32768 waves
        const int rows_per_wave = (n_upd + n_waves - 1) / n_waves;
        scatter_add<<<blocks, tpb, 0, stream>>>(index, B, out, n_upd, rows_per_wave);
    }
}